// CGTA_34608846471843
// MI455X (gfx1250) — compile-verified
//
#include <hip/hip_runtime.h>
#include <math.h>

typedef __attribute__((ext_vector_type(2))) float v2f;
typedef __attribute__((ext_vector_type(8))) float v8f;

#define WMMA_F32(a, b, c) \
  __builtin_amdgcn_wmma_f32_16x16x4_f32(false, (a), false, (b), (short)0, (c), false, false)

// (DH * 0.5)^-0.5 = 32^-0.5
#define LSCALE 0.17677669529663687f

// Raw 32-bit LDS byte offset of a __shared__ object (ptrtoint of addrspace(3)).
__device__ __forceinline__ unsigned lds_addr_of(const void* p) {
  return (unsigned)(unsigned long long)(const __attribute__((address_space(3))) char*)p;
}

// CDNA5 async DMA: 16 bytes global -> LDS per lane, tracked by ASYNCcnt.
__device__ __forceinline__ void async_copy_b128(const float* g, const float* lds_dst) {
  asm volatile("global_load_async_to_lds_b128 %0, %1, off"
               :: "v"(lds_addr_of(lds_dst)), "v"((unsigned long long)g)
               : "memory");
}

#define WAIT_ASYNC(imm) asm volatile("s_wait_asynccnt " #imm ::: "memory")

// ---------------------------------------------------------------------------
// Generic fp32 WMMA GEMM: C[M,N] = A[M,K] @ W[N,K]^T (+bias) (opt relu)
// Tile: BM=128 (8 waves x 16 rows), BN=64 (4 accumulators), KC=32.
// Double-buffered LDS filled by global_load_async_to_lds_b128 (no VGPR staging);
// prefetch of chunk c+1 overlaps WMMA on chunk c. Stride-36 padding keeps the
// b64 fragment reads bank-conflict free and async b128 stores 16B aligned.
// Requires M%128==0, N%64==0, K%32==0 (true for all call sites here).
// ---------------------------------------------------------------------------
__global__ __launch_bounds__(256) void gemm_f32(
    const float* __restrict__ A, const float* __restrict__ W,
    const float* __restrict__ bias, float* __restrict__ C,
    int M, int N, int K, int relu)
{
  __shared__ float As[2][128][36];
  __shared__ float Ws[2][64][36];
  const int tid = threadIdx.x;
  const int wid = tid >> 5, lane = tid & 31;
  const int hlf = lane >> 4, mr = lane & 15;
  const int rowBase = blockIdx.x * 128;
  const int colBase = blockIdx.y * 64;
  (void)M;

  // 6 async b128 per thread per K-chunk: 4 for A (128x32), 2 for W (64x32)
  auto issue = [&](int bf, int k0) {
    #pragma unroll
    for (int i = 0; i < 4; ++i) {
      int idx = tid + i * 256;
      int r = idx >> 3, c4 = (idx & 7) << 2;
      async_copy_b128(A + (size_t)(rowBase + r) * K + k0 + c4, &As[bf][r][c4]);
    }
    #pragma unroll
    for (int i = 0; i < 2; ++i) {
      int idx = tid + i * 256;
      int r = idx >> 3, c4 = (idx & 7) << 2;
      async_copy_b128(W + (size_t)(colBase + r) * K + k0 + c4, &Ws[bf][r][c4]);
    }
  };

  issue(0, 0);
  v8f acc[4] = {};
  const int nchunks = K >> 5;
  for (int c = 0; c < nchunks; ++c) {
    const int cur = c & 1;
    if (c + 1 < nchunks) {
      issue(cur ^ 1, (c + 1) << 5);     // prefetch next chunk into other buffer
      WAIT_ASYNC(0x6);                  // first 6 (current buffer) have landed
    } else {
      WAIT_ASYNC(0x0);
    }
    __syncthreads();                    // all waves' async writes visible
    #pragma unroll
    for (int kk = 0; kk < 32; kk += 4) {
      v2f a;
      a.x = As[cur][wid * 16 + mr][kk + hlf * 2];
      a.y = As[cur][wid * 16 + mr][kk + hlf * 2 + 1];
      #pragma unroll
      for (int nt = 0; nt < 4; ++nt) {
        v2f bb;
        bb.x = Ws[cur][nt * 16 + mr][kk + hlf * 2];
        bb.y = Ws[cur][nt * 16 + mr][kk + hlf * 2 + 1];
        acc[nt] = WMMA_F32(a, bb, acc[nt]);
      }
    }
    __syncthreads();                    // done reading before buffer is refilled
  }
  #pragma unroll
  for (int nt = 0; nt < 4; ++nt) {
    int n = colBase + nt * 16 + mr;
    float bv = bias ? bias[n] : 0.0f;
    #pragma unroll
    for (int i = 0; i < 8; ++i) {
      int m = rowBase + wid * 16 + hlf * 8 + i;
      float v = acc[nt][i] + bv;
      if (relu) v = fmaxf(v, 0.0f);
      C[(size_t)m * N + n] = v;
    }
  }
}

// ---------------------------------------------------------------------------
// Curvature: depthwise 3x3 conv + channel mean. One wave per pixel.
// ---------------------------------------------------------------------------
__global__ __launch_bounds__(256) void curv_kernel(
    const float* __restrict__ x, const float* __restrict__ cw, float* __restrict__ curv)
{
  const int wid = threadIdx.x >> 5, lane = threadIdx.x & 31;
  const int p = blockIdx.x * 8 + wid;          // 0..131071 = b*16384 + n
  const int b = p >> 14, n = p & 16383;
  const int py = n >> 7, px = n & 127;
  float acc = 0.0f;
  #pragma unroll
  for (int t = 0; t < 9; ++t) {
    const int dy = t / 3 - 1, dx = t % 3 - 1;
    const int yy = py + dy, xx = px + dx;
    if (yy >= 0 && yy < 128 && xx >= 0 && xx < 128) {
      const float* xp = x + ((size_t)b * 16384 + yy * 128 + xx) * 512;
      #pragma unroll
      for (int j = 0; j < 16; ++j) {
        int c = lane + 32 * j;                  // coalesced across lanes
        acc += xp[c] * cw[c * 9 + t];
      }
    }
  }
  #pragma unroll
  for (int off = 16; off > 0; off >>= 1) acc += __shfl_xor(acc, off, 32);
  if (lane == 0) curv[p] = acc * (1.0f / 512.0f);
}

// Per-batch LayerNorm (no affine) over 16384 values, two-pass.
__global__ __launch_bounds__(1024) void ln_curv_kernel(float* __restrict__ curv)
{
  const int b = blockIdx.x;
  float* p = curv + (size_t)b * 16384;
  const int tid = threadIdx.x, wid = tid >> 5, lane = tid & 31;
  __shared__ float red[32];
  __shared__ float s_mean, s_rstd;
  float s = 0.0f;
  for (int i = tid; i < 16384; i += 1024) s += p[i];
  #pragma unroll
  for (int off = 16; off > 0; off >>= 1) s += __shfl_xor(s, off, 32);
  if (lane == 0) red[wid] = s;
  __syncthreads();
  if (wid == 0) {
    float t = red[lane];
    #pragma unroll
    for (int off = 16; off > 0; off >>= 1) t += __shfl_xor(t, off, 32);
    if (lane == 0) s_mean = t * (1.0f / 16384.0f);
  }
  __syncthreads();
  const float mean = s_mean;
  float v = 0.0f;
  for (int i = tid; i < 16384; i += 1024) { float d = p[i] - mean; v += d * d; }
  #pragma unroll
  for (int off = 16; off > 0; off >>= 1) v += __shfl_xor(v, off, 32);
  if (lane == 0) red[wid] = v;
  __syncthreads();
  if (wid == 0) {
    float t = red[lane];
    #pragma unroll
    for (int off = 16; off > 0; off >>= 1) t += __shfl_xor(t, off, 32);
    if (lane == 0) s_rstd = rsqrtf(t * (1.0f / 16384.0f) + 1e-5f);
  }
  __syncthreads();
  const float rstd = s_rstd;
  for (int i = tid; i < 16384; i += 1024) p[i] = (p[i] - mean) * rstd;
}

// gate = sigmoid(hidden . w2 + b2); score = 0.5*(|curv| + gate). Wave per token.
__global__ __launch_bounds__(256) void gate_score_kernel(
    const float* __restrict__ hidden, const float* __restrict__ w2,
    const float* __restrict__ b2, const float* __restrict__ curv,
    float* __restrict__ score)
{
  const int wid = threadIdx.x >> 5, lane = threadIdx.x & 31;
  const int p = blockIdx.x * 8 + wid;
  const float* hp = hidden + (size_t)p * 256;
  float acc = 0.0f;
  #pragma unroll
  for (int j = 0; j < 8; ++j) { int c = lane + 32 * j; acc += hp[c] * w2[c]; }
  #pragma unroll
  for (int off = 16; off > 0; off >>= 1) acc += __shfl_xor(acc, off, 32);
  if (lane == 0) {
    float g = 1.0f / (1.0f + expf(-(acc + b2[0])));
    score[p] = 0.5f * (fabsf(curv[p]) + g);
  }
}

// Exact jax.lax.top_k semantics (desc value, ties -> lower index first).
// Scores are strictly positive -> float bits are order-monotone as uint.
__global__ __launch_bounds__(1024) void topk_kernel(
    const float* __restrict__ score, const float* __restrict__ curv,
    int* __restrict__ idx, float* __restrict__ stk, float* __restrict__ ctk)
{
  const int b = blockIdx.x;
  const int tid = threadIdx.x, wid = tid >> 5, lane = tid & 31;
  unsigned long long keys[16];
  #pragma unroll
  for (int j = 0; j < 16; ++j) {
    int n = tid + 1024 * j;
    unsigned int fb = __float_as_uint(score[(size_t)b * 16384 + n]);
    keys[j] = ((unsigned long long)fb << 32) |
              (unsigned long long)(0xFFFFFFFFu - (unsigned int)n);
  }
  __shared__ unsigned long long wred[32];
  __shared__ unsigned long long s_best;
  for (int r = 0; r < 64; ++r) {
    unsigned long long k = 0ull;
    #pragma unroll
    for (int j = 0; j < 16; ++j) k = keys[j] > k ? keys[j] : k;
    #pragma unroll
    for (int off = 16; off > 0; off >>= 1) {
      unsigned long long o = __shfl_xor(k, off, 32);
      k = o > k ? o : k;
    }
    if (lane == 0) wred[wid] = k;
    __syncthreads();
    if (wid == 0) {
      unsigned long long t = wred[lane];
      #pragma unroll
      for (int off = 16; off > 0; off >>= 1) {
        unsigned long long o = __shfl_xor(t, off, 32);
        t = o > t ? o : t;
      }
      if (lane == 0) s_best = t;
    }
    __syncthreads();
    const unsigned long long best = s_best;
    const unsigned int bi = 0xFFFFFFFFu - (unsigned int)(best & 0xFFFFFFFFull);
    if ((bi & 1023u) == (unsigned)tid) keys[bi >> 10] = 0ull;   // retire winner
    if (tid == 0) {
      idx[b * 64 + r] = (int)bi;
      stk[b * 64 + r] = __uint_as_float((unsigned int)(best >> 32));
      ctk[b * 64 + r] = curv[(size_t)b * 16384 + bi];
    }
    __syncthreads();
  }
}

__global__ __launch_bounds__(128) void gather_kernel(
    const float* __restrict__ x, const int* __restrict__ idx, float* __restrict__ xt)
{
  const int bt = blockIdx.x;                    // b*64 + t
  const int b = bt >> 6;
  const int src = idx[bt];
  const int c4 = threadIdx.x * 4;
  float4 v = *(const float4*)(x + ((size_t)b * 16384 + src) * 512 + c4);
  *(float4*)(xt + (size_t)bt * 512 + c4) = v;
}

// LayerNorm(affine) + exact GELU, in place. Wave per row of 256.
__global__ __launch_bounds__(256) void ln_gelu_kernel(
    float* __restrict__ kv, const float* __restrict__ g, const float* __restrict__ bb)
{
  const int wid = threadIdx.x >> 5, lane = threadIdx.x & 31;
  const int row = blockIdx.x * 8 + wid;
  float* p = kv + (size_t)row * 256;
  float v[8];
  #pragma unroll
  for (int j = 0; j < 8; ++j) v[j] = p[lane + 32 * j];
  float s = 0.0f;
  #pragma unroll
  for (int j = 0; j < 8; ++j) s += v[j];
  #pragma unroll
  for (int off = 16; off > 0; off >>= 1) s += __shfl_xor(s, off, 32);
  const float mean = s * (1.0f / 256.0f);
  float vv = 0.0f;
  #pragma unroll
  for (int j = 0; j < 8; ++j) { float d = v[j] - mean; vv += d * d; }
  #pragma unroll
  for (int off = 16; off > 0; off >>= 1) vv += __shfl_xor(vv, off, 32);
  const float rstd = rsqrtf(vv * (1.0f / 256.0f) + 1e-5f);
  #pragma unroll
  for (int j = 0; j < 8; ++j) {
    int c = lane + 32 * j;
    float y = (v[j] - mean) * rstd * g[c] + bb[c];
    p[c] = 0.5f * y * (1.0f + erff(y * 0.70710678118654752f));
  }
}

// v_final = v*score + dwconv3x3(v*score, cpe_w, cpe_b) on the 8x8 top-k grid.
__global__ __launch_bounds__(128) void cpe_kernel(
    const float* __restrict__ vbuf, const float* __restrict__ stk,
    const float* __restrict__ cw, const float* __restrict__ cb,
    float* __restrict__ vfin)
{
  const int bt = blockIdx.x;
  const int b = bt >> 6, t = bt & 63;
  const int gy = t >> 3, gx = t & 7;
  for (int c = threadIdx.x; c < 512; c += 128) {
    float acc = cb[c];
    #pragma unroll
    for (int tap = 0; tap < 9; ++tap) {
      int dy = tap / 3 - 1, dx = tap % 3 - 1;
      int yy = gy + dy, xx = gx + dx;
      if (yy >= 0 && yy < 8 && xx >= 0 && xx < 8) {
        int tn = b * 64 + yy * 8 + xx;
        acc += cw[c * 9 + tap] * vbuf[(size_t)tn * 512 + c] * stk[tn];
      }
    }
    vfin[(size_t)bt * 512 + c] = vbuf[(size_t)bt * 512 + c] * stk[bt] + acc;
  }
}

// ---------------------------------------------------------------------------
// Fused attention: per (64-token tile, head, batch) block of 4 waves.
// Q/K tiles arrive via async global->LDS DMA; V^T needs a transpose so it goes
// through VGPRs. logits = Q K^T * SCALE (WMMA), curvature-modulated dual
// softmax in registers, attn spilled per-wave to LDS, AV as out^T = V^T@attn^T.
// ---------------------------------------------------------------------------
__global__ __launch_bounds__(128) void attn_kernel(
    const float* __restrict__ qbuf, const float* __restrict__ kbuf,
    const float* __restrict__ vfin, const float* __restrict__ ctk,
    const float* __restrict__ alpha_p, const float* __restrict__ beta_p,
    float* __restrict__ out)
{
  __shared__ float Qs[64][36];       // q tile [token][dq]
  __shared__ float Ks[64][36];       // k tile [kt][dq]
  __shared__ float Vt[64][68];       // v^T tile [d][kt]
  __shared__ float At[4][16][66];    // per-wave attn spill [m][kt]
  const int tid = threadIdx.x, wid = tid >> 5, lane = tid & 31;
  const int hlf = lane >> 4, mr = lane & 15;
  const int rb = blockIdx.x * 64;
  const int h = blockIdx.y, b = blockIdx.z;

  #pragma unroll
  for (int i = 0; i < 4; ++i) {      // Q: 64x32, async DMA to LDS
    int idx = tid + i * 128;
    int r = idx >> 3, c4 = (idx & 7) << 2;
    async_copy_b128(qbuf + ((size_t)b * 16384 + rb + r) * 256 + h * 32 + c4, &Qs[r][c4]);
  }
  #pragma unroll
  for (int i = 0; i < 4; ++i) {      // K: 64x32, async DMA to LDS
    int idx = tid + i * 128;
    int r = idx >> 3, c4 = (idx & 7) << 2;
    async_copy_b128(kbuf + ((size_t)b * 64 + r) * 256 + h * 32 + c4, &Ks[r][c4]);
  }
  #pragma unroll
  for (int i = 0; i < 8; ++i) {      // V^T: 64x64 transposed into LDS via VGPRs
    int idx = tid + i * 128;
    int n = idx >> 4, d4 = (idx & 15) << 2;
    float4 v = *(const float4*)(vfin + ((size_t)b * 64 + n) * 512 + h * 64 + d4);
    Vt[d4][n] = v.x; Vt[d4 + 1][n] = v.y; Vt[d4 + 2][n] = v.z; Vt[d4 + 3][n] = v.w;
  }
  WAIT_ASYNC(0x0);
  __syncthreads();

  // logits tile [16 tokens x 64 kt]
  v8f acc[4] = {};
  #pragma unroll
  for (int kk = 0; kk < 32; kk += 4) {
    v2f a;
    a.x = Qs[wid * 16 + mr][kk + hlf * 2];
    a.y = Qs[wid * 16 + mr][kk + hlf * 2 + 1];
    #pragma unroll
    for (int nt = 0; nt < 4; ++nt) {
      v2f bb;
      bb.x = Ks[nt * 16 + mr][kk + hlf * 2];
      bb.y = Ks[nt * 16 + mr][kk + hlf * 2 + 1];
      acc[nt] = WMMA_F32(a, bb, acc[nt]);
    }
  }

  const float alpha = *alpha_p, beta = *beta_p;
  float ct[4];
  #pragma unroll
  for (int nt = 0; nt < 4; ++nt) ct[nt] = ctk[b * 64 + nt * 16 + mr];

  float m1[8], m2[8], s1[8], s2[8];
  #pragma unroll
  for (int i = 0; i < 8; ++i) { m1[i] = -1e30f; m2[i] = -1e30f; s1[i] = 0.0f; s2[i] = 0.0f; }
  #pragma unroll
  for (int nt = 0; nt < 4; ++nt)
    #pragma unroll
    for (int i = 0; i < 8; ++i) {
      float lg = acc[nt][i] * LSCALE;
      acc[nt][i] = lg;
      float md = lg * (1.0f + alpha * ct[nt]);
      m1[i] = fmaxf(m1[i], lg); m2[i] = fmaxf(m2[i], md);
    }
  #pragma unroll
  for (int off = 1; off < 16; off <<= 1)     // reduce over kt within 16-lane half
    #pragma unroll
    for (int i = 0; i < 8; ++i) {
      m1[i] = fmaxf(m1[i], __shfl_xor(m1[i], off, 32));
      m2[i] = fmaxf(m2[i], __shfl_xor(m2[i], off, 32));
    }
  #pragma unroll
  for (int nt = 0; nt < 4; ++nt)
    #pragma unroll
    for (int i = 0; i < 8; ++i) {
      float lg = acc[nt][i];
      float md = lg * (1.0f + alpha * ct[nt]);
      s1[i] += __expf(lg - m1[i]); s2[i] += __expf(md - m2[i]);
    }
  #pragma unroll
  for (int off = 1; off < 16; off <<= 1)
    #pragma unroll
    for (int i = 0; i < 8; ++i) {
      s1[i] += __shfl_xor(s1[i], off, 32);
      s2[i] += __shfl_xor(s2[i], off, 32);
    }
  #pragma unroll
  for (int i = 0; i < 8; ++i) { s1[i] = 1.0f / s1[i]; s2[i] = 1.0f / s2[i]; }
  #pragma unroll
  for (int nt = 0; nt < 4; ++nt)
    #pragma unroll
    for (int i = 0; i < 8; ++i) {
      float lg = acc[nt][i];
      float md = lg * (1.0f + alpha * ct[nt]);
      float pr = beta * __expf(lg - m1[i]) * s1[i] +
                 (1.0f - beta) * __expf(md - m2[i]) * s2[i];
      At[wid][hlf * 8 + i][nt * 16 + mr] = pr;
    }
  __syncthreads();

  // out^T[d][m] = sum_n V^T[d][n] * attn[m][n]
  v8f oacc[4] = {};
  #pragma unroll
  for (int kk = 0; kk < 64; kk += 4) {
    v2f bb;                                   // B = attn^T : col m = mr
    bb.x = At[wid][mr][kk + hlf * 2];
    bb.y = At[wid][mr][kk + hlf * 2 + 1];
    #pragma unroll
    for (int dt = 0; dt < 4; ++dt) {
      v2f a;                                  // A = V^T : row d
      a.x = Vt[dt * 16 + mr][kk + hlf * 2];
      a.y = Vt[dt * 16 + mr][kk + hlf * 2 + 1];
      oacc[dt] = WMMA_F32(a, bb, oacc[dt]);
    }
  }
  const int tok = rb + wid * 16 + mr;
  #pragma unroll
  for (int dt = 0; dt < 4; ++dt)
    #pragma unroll
    for (int i = 0; i < 8; ++i) {
      int d = dt * 16 + hlf * 8 + i;
      out[((size_t)b * 16384 + tok) * 512 + h * 64 + d] = oacc[dt][i];
    }
}

// ---------------------------------------------------------------------------
extern "C" void kernel_launch(void* const* d_in, const int* in_sizes, int n_in,
                              void* d_out, int out_size, void* d_ws, size_t ws_size,
                              hipStream_t stream)
{
  (void)in_sizes; (void)n_in; (void)out_size; (void)ws_size;
  const float* x      = (const float*)d_in[0];
  const float* curv_w = (const float*)d_in[1];
  const float* gw1    = (const float*)d_in[2];
  const float* gb1    = (const float*)d_in[3];
  const float* gw2    = (const float*)d_in[4];
  const float* gb2    = (const float*)d_in[5];
  const float* q_w    = (const float*)d_in[6];
  const float* kvr_w  = (const float*)d_in[7];
  const float* kvr_b  = (const float*)d_in[8];
  const float* ln_g   = (const float*)d_in[9];
  const float* ln_b   = (const float*)d_in[10];
  const float* k_w    = (const float*)d_in[11];
  const float* v_w    = (const float*)d_in[12];
  const float* cpe_w  = (const float*)d_in[13];
  const float* cpe_b  = (const float*)d_in[14];
  const float* proj_w = (const float*)d_in[15];
  const float* proj_b = (const float*)d_in[16];
  const float* alpha  = (const float*)d_in[17];
  const float* beta   = (const float*)d_in[18];
  float* out = (float*)d_out;

  // Workspace layout (floats). Peak ~408 MB.
  float* ws    = (float*)d_ws;
  float* curv  = ws;                               // 131072
  float* score = curv + 131072;                    // 131072
  float* stk   = score + 131072;                   // 512
  float* ctk   = stk + 512;                        // 512
  int*   idx   = (int*)(ctk + 512);                // 512 ints
  float* xt    = ctk + 1024;                       // 512*512
  float* kv    = xt + 512 * 512;                   // 512*256
  float* kb    = kv + 512 * 256;                   // 512*256
  float* vb    = kb + 512 * 256;                   // 512*512
  float* vf    = vb + 512 * 512;                   // 512*512
  float* big   = vf + 512 * 512;                   // 131072*512: hidden, later attn_out
  float* qb    = big + (size_t)131072 * 512;       // 131072*256
  float* hidden   = big;
  float* attn_out = big;

  curv_kernel<<<16384, 256, 0, stream>>>(x, curv_w, curv);
  ln_curv_kernel<<<8, 1024, 0, stream>>>(curv);
  // hidden = relu(x @ gate_w1^T + gate_b1)
  gemm_f32<<<dim3(1024, 4), 256, 0, stream>>>(x, gw1, gb1, hidden, 131072, 256, 512, 1);
  gate_score_kernel<<<16384, 256, 0, stream>>>(hidden, gw2, gb2, curv, score);
  topk_kernel<<<8, 1024, 0, stream>>>(score, curv, idx, stk, ctk);
  gather_kernel<<<512, 128, 0, stream>>>(x, idx, xt);
  // q = x @ q_w^T
  gemm_f32<<<dim3(1024, 4), 256, 0, stream>>>(x, q_w, nullptr, qb, 131072, 256, 512, 0);
  // kv = gelu(LN(xt @ kvr_w^T + kvr_b))
  gemm_f32<<<dim3(4, 4), 256, 0, stream>>>(xt, kvr_w, kvr_b, kv, 512, 256, 512, 0);
  ln_gelu_kernel<<<64, 256, 0, stream>>>(kv, ln_g, ln_b);
  // k = kv @ k_w^T ; v = kv @ v_w^T
  gemm_f32<<<dim3(4, 4), 256, 0, stream>>>(kv, k_w, nullptr, kb, 512, 256, 256, 0);
  gemm_f32<<<dim3(4, 8), 256, 0, stream>>>(kv, v_w, nullptr, vb, 512, 512, 256, 0);
  cpe_kernel<<<512, 128, 0, stream>>>(vb, stk, cpe_w, cpe_b, vf);
  attn_kernel<<<dim3(256, 8, 8), 128, 0, stream>>>(qb, kb, vf, ctk, alpha, beta, attn_out);
  // out = attn_out @ proj_w^T + proj_b
  gemm_f32<<<dim3(1024, 8), 256, 0, stream>>>(attn_out, proj_w, proj_b, out, 131072, 512, 512, 0);
}